// GCN_TCN_72902774882834
// MI455X (gfx1250) — compile-verified
//
#include <hip/hip_runtime.h>
#include <hip/hip_bf16.h>

typedef float v2f __attribute__((ext_vector_type(2)));
typedef float v4f __attribute__((ext_vector_type(4)));
typedef float v8f __attribute__((ext_vector_type(8)));
typedef int   v4i __attribute__((ext_vector_type(4)));

typedef __attribute__((address_space(1))) v4i* glb_v4i_p;
typedef __attribute__((address_space(3))) v4i* lds_v4i_p;

#define NODES 30
#define MTOT  61440   // B*30
#define BATCH 2048
#define ECNT  491520

#define USE_ASYNC_LDS (__has_builtin(__builtin_amdgcn_global_load_async_to_lds_b128) && \
                       __has_builtin(__builtin_amdgcn_s_wait_asynccnt))

// ---------------------------------------------------------------------------
// f32 WMMA 16x16x4: D = A(16x4) * B(4x16) + C   (wave32, full EXEC)
// ---------------------------------------------------------------------------
__device__ __forceinline__ v8f wmma_f32_k4(v2f a, v2f b, v8f c) {
  return __builtin_amdgcn_wmma_f32_16x16x4_f32(
      false, a, false, b, (short)0, c, false, false);
}

// 16B global -> LDS stage: async DMA if available, else load+ds_store
__device__ __forceinline__ void stage16(const float* __restrict__ src, float* dst) {
#if USE_ASYNC_LDS
  __builtin_amdgcn_global_load_async_to_lds_b128((glb_v4i_p)src, (lds_v4i_p)dst, 0, 0);
#else
  *(v4f*)dst = *(const v4f*)src;
#endif
}
__device__ __forceinline__ void stage_fence() {
#if USE_ASYNC_LDS
  __builtin_amdgcn_s_wait_asynccnt(0);
#endif
}

// ---------------------------------------------------------------------------
// GCN stage
// ---------------------------------------------------------------------------
__global__ void gcn_deg_init(float* __restrict__ deg) {
  int i = blockIdx.x * blockDim.x + threadIdx.x;
  if (i < MTOT) deg[i] = 1.0f;  // self-loop contribution
}

__global__ void gcn_deg_count(const long long* __restrict__ ei, float* __restrict__ deg) {
  int e = blockIdx.x * blockDim.x + threadIdx.x;
  if (e < ECNT) atomicAdd(&deg[(int)ei[ECNT + e]], 1.0f);  // dst row
}

__global__ void gcn_dinv(float* __restrict__ deg) {
  int i = blockIdx.x * blockDim.x + threadIdx.x;
  if (i < MTOT) deg[i] = rsqrtf(deg[i]);  // deg >= 1 always
}

__global__ void gcn_h_kernel(const float* __restrict__ x, const float* __restrict__ w,
                             float* __restrict__ h) {
  int nidx = blockIdx.x * blockDim.x + threadIdx.x;
  if (nidx >= MTOT) return;
  float xr[12];
#pragma unroll
  for (int i = 0; i < 12; ++i) xr[i] = x[(size_t)nidx * 12 + i];
#pragma unroll
  for (int o = 0; o < 12; ++o) {
    float s = 0.f;
#pragma unroll
    for (int i = 0; i < 12; ++i) s = fmaf(xr[i], w[o * 12 + i], s);
    h[(size_t)nidx * 12 + o] = s;
  }
}

__global__ void gcn_self(const float* __restrict__ h, const float* __restrict__ dinv,
                         const float* __restrict__ bias, float* __restrict__ X0) {
  int idx = blockIdx.x * blockDim.x + threadIdx.x;
  if (idx >= MTOT * 12) return;
  int nd = idx / 12, c = idx % 12;
  float di = dinv[nd];
  X0[idx] = di * di * h[idx] + bias[c];   // full overwrite: poison-safe
}

__global__ void gcn_scatter(const long long* __restrict__ ei, const float* __restrict__ h,
                            const float* __restrict__ dinv, float* __restrict__ X0) {
  long long idx = (long long)blockIdx.x * blockDim.x + threadIdx.x;
  if (idx >= (long long)ECNT * 12) return;
  int e = (int)(idx / 12);
  int c = (int)(idx % 12);
  int s = (int)ei[e];
  int d = (int)ei[ECNT + e];
  float nrm = dinv[s] * dinv[d];
  atomicAdd(&X0[(size_t)d * 12 + c], nrm * h[(size_t)s * 12 + c]);
}

// ---------------------------------------------------------------------------
// Weight pre-transposes
// ---------------------------------------------------------------------------
__global__ void transpose_conv_w(const float* __restrict__ w, float* __restrict__ wt,
                                 int O, int I) {  // [O][I][3] -> [3][I][O]
  int idx = blockIdx.x * blockDim.x + threadIdx.x;
  if (idx >= O * I * 3) return;
  int o = idx / (I * 3);
  int rem = idx % (I * 3);
  int c = rem / 3, kk = rem % 3;
  wt[((size_t)kk * I + c) * O + o] = w[idx];
}

__global__ void transpose_down_w(const float* __restrict__ w, float* __restrict__ wt,
                                 int O, int I) {  // [O][I] -> [I][O]
  int idx = blockIdx.x * blockDim.x + threadIdx.x;
  if (idx >= O * I) return;
  int o = idx / I, c = idx % I;
  wt[(size_t)c * O + o] = w[idx];
}

__global__ void transpose_fc_w(const float* __restrict__ w, float* __restrict__ wt) {
  // fc_w [72][3840] (k = c*30+t)  ->  wt [3840 (k' = t*128+c)][80], cols 72..79 = 0
  int idx = blockIdx.x * blockDim.x + threadIdx.x;
  if (idx >= 3840 * 80) return;
  int kp = idx / 80, o = idx % 80;
  int t = kp / 128, c = kp % 128;
  wt[idx] = (o < 72) ? w[(size_t)o * 3840 + c * 30 + t] : 0.f;
}

// ---------------------------------------------------------------------------
// Fused TemporalBlock: Y = relu( relu(conv_dil(X)+cb) + down1x1(X)+db )
// Block = 128 threads (4 waves): 64-row M supertile, wave w -> N tile.
// Per 32-wide K chunk: stage (64+2*DIL)-row activation panel into LDS
// (async DMA), then all 3 conv taps + downsample read shifted panel rows.
// Causal zero-padding = hoisted per-lane base-offset select into an LDS
// zero row; inner loop is pure base+immediate DS/VMEM addressing + WMMA.
// ---------------------------------------------------------------------------
template <int CIN, int COUT, int DIL>
__global__ __launch_bounds__(128)
void tcn_block(const float* __restrict__ X,    // [MTOT][CIN]  ([b][t][c])
               const float* __restrict__ WcT,  // [3][CIN][COUT]
               const float* __restrict__ cb,
               const float* __restrict__ WdT,  // [CIN][COUT]
               const float* __restrict__ db,
               float* __restrict__ Y)          // [MTOT][COUT]
{
  constexpr int KCH   = (CIN < 32) ? CIN : 32;          // K chunk width
  constexpr int PITCH = (KCH % 8 == 0) ? KCH + 4 : KCH + 8; // bank-conflict-free, 16B-aligned
  constexpr int PROWS = 64 + 2 * DIL;                   // staged rows
  constexpr int ZOFF  = PROWS * PITCH;                  // LDS zero row
  constexpr int SEGS  = KCH / 4;                        // b128 segs per row
  __shared__ float sm[ZOFF + KCH];

  const int tid  = threadIdx.x;
  const int lane = tid & 31;
  const int half = lane >> 4;          // K-offset selector (A/B fragments)
  const int l15  = lane & 15;
  const int wave = tid >> 5;
  const int m0   = blockIdx.x * 64;
  const int n    = (blockIdx.y * 4 + wave) * 16 + l15;

  // Hoisted per-mt LDS element offsets (2*half K-phase folded in); the
  // causal-pad select happens ONCE here, not in the inner loop.
  int off0[4], off1[4], off2[4];
#pragma unroll
  for (int mt = 0; mt < 4; ++mt) {
    const int pb = mt * 16 + l15;
    const int tm = (m0 + pb) % NODES;
    off2[mt] = (pb + 2 * DIL) * PITCH + 2 * half;                      // always valid
    off0[mt] = ((tm >= 2 * DIL) ? pb * PITCH : ZOFF) + 2 * half;       // tap0 pad
    off1[mt] = ((tm >= DIL) ? (pb + DIL) * PITCH : ZOFF) + 2 * half;   // tap1 pad
  }

  // Pre-biased weight pointers (lane column + K-phase); advanced per chunk.
  const float* pD  = WdT + (size_t)(2 * half) * COUT + n;
  const float* pC0 = WcT + (size_t)(2 * half) * COUT + n;
  const float* pC1 = pC0 + (size_t)CIN * COUT;
  const float* pC2 = pC1 + (size_t)CIN * COUT;

  v8f accC[4], accD[4];
#pragma unroll
  for (int mt = 0; mt < 4; ++mt) {
    accC[mt] = (v8f){0.f, 0.f, 0.f, 0.f, 0.f, 0.f, 0.f, 0.f};
    accD[mt] = (v8f){0.f, 0.f, 0.f, 0.f, 0.f, 0.f, 0.f, 0.f};
  }

  if (tid < KCH) sm[ZOFF + tid] = 0.f;   // zero row (causal padding source)

  for (int k0 = 0; k0 < CIN; k0 += KCH) {
    __syncthreads();                     // prior reads done / zero row ready
    for (int g = tid; g < PROWS * SEGS; g += 128) {
      int p = g / SEGS, seg = g % SEGS;
      int r = m0 - 2 * DIL + p;          // r < 0 rows are never read validly
      if (r >= 0)
        stage16(X + (size_t)r * CIN + k0 + seg * 4, &sm[p * PITCH + seg * 4]);
    }
    stage_fence();
    __syncthreads();

#pragma unroll
    for (int ks = 0; ks < KCH; ks += 4) {  // ks is a literal: imm offsets below
      v2f bD, bC0, bC1, bC2;
      bD.x  = pD [(size_t)ks * COUT];       bD.y  = pD [(size_t)(ks + 1) * COUT];
      bC0.x = pC0[(size_t)ks * COUT];       bC0.y = pC0[(size_t)(ks + 1) * COUT];
      bC1.x = pC1[(size_t)ks * COUT];       bC1.y = pC1[(size_t)(ks + 1) * COUT];
      bC2.x = pC2[(size_t)ks * COUT];       bC2.y = pC2[(size_t)(ks + 1) * COUT];
#pragma unroll
      for (int mt = 0; mt < 4; ++mt) {
        // tap2 (t-0): fragment shared by conv tap2 and downsample
        v2f a2 = *(const v2f*)&sm[off2[mt] + ks];
        accD[mt] = wmma_f32_k4(a2, bD,  accD[mt]);
        accC[mt] = wmma_f32_k4(a2, bC2, accC[mt]);
        v2f a0 = *(const v2f*)&sm[off0[mt] + ks];
        accC[mt] = wmma_f32_k4(a0, bC0, accC[mt]);
        v2f a1 = *(const v2f*)&sm[off1[mt] + ks];
        accC[mt] = wmma_f32_k4(a1, bC1, accC[mt]);
      }
    }
    pD  += (size_t)KCH * COUT;
    pC0 += (size_t)KCH * COUT;
    pC1 += (size_t)KCH * COUT;
    pC2 += (size_t)KCH * COUT;
  }

  const float cbn = cb[n];
  const float dbn = db[n];
#pragma unroll
  for (int mt = 0; mt < 4; ++mt) {
#pragma unroll
    for (int r = 0; r < 8; ++r) {
      const int row = m0 + mt * 16 + r + 8 * half;  // D layout: M = r + 8*half
      float oc = fmaxf(accC[mt][r] + cbn, 0.f);
      float o  = fmaxf(oc + accD[mt][r] + dbn, 0.f);
      Y[(size_t)row * COUT + n] = o;
    }
  }
}

// ---------------------------------------------------------------------------
// FC: out[2048,72] = X3.view(2048,3840) @ WT(3840x80, zero-padded) + b
// ---------------------------------------------------------------------------
template <int WAVES>
__global__ __launch_bounds__(WAVES * 32)
void fc_kernel(const float* __restrict__ X,    // [BATCH][3840]  (t*128+c order)
               const float* __restrict__ WT,   // [3840][80]
               const float* __restrict__ bias, // [72]
               float* __restrict__ out)        // [BATCH][72]
{
  const int lane = threadIdx.x & 31;
  const int half = lane >> 4;
  const int l15  = lane & 15;
  const int wave = threadIdx.x >> 5;
  const int m = blockIdx.x * 16 + l15;
  const int n = (blockIdx.y * WAVES + wave) * 16 + l15;  // 0..79

  v8f acc = {0.f, 0.f, 0.f, 0.f, 0.f, 0.f, 0.f, 0.f};
  const float* xk = X + (size_t)m * 3840 + 2 * half;     // pre-biased
  const float* wk = WT + (size_t)(2 * half) * 80 + n;    // pre-biased
#pragma unroll 8
  for (int k0 = 0; k0 < 3840; k0 += 4) {
    v2f a; a.x = xk[k0];                a.y = xk[k0 + 1];
    v2f w; w.x = wk[(size_t)k0 * 80];   w.y = wk[(size_t)(k0 + 1) * 80];
    acc = wmma_f32_k4(a, w, acc);
  }
  float bn = (n < 72) ? bias[n] : 0.f;
#pragma unroll
  for (int r = 0; r < 8; ++r) {
    const int row = blockIdx.x * 16 + r + 8 * half;
    if (n < 72) out[(size_t)row * 72 + n] = acc[r] + bn;
  }
}

// ---------------------------------------------------------------------------
// Host launcher
// ---------------------------------------------------------------------------
extern "C" void kernel_launch(void* const* d_in, const int* in_sizes, int n_in,
                              void* d_out, int out_size, void* d_ws, size_t ws_size,
                              hipStream_t stream) {
  (void)in_sizes; (void)n_in; (void)out_size; (void)ws_size;
  const float*     x      = (const float*)d_in[0];
  const long long* ei     = (const long long*)d_in[1];   // int64 [2][E]
  const float*     gcn_w  = (const float*)d_in[2];
  const float*     gcn_b  = (const float*)d_in[3];
  const float*     cw0 = (const float*)d_in[4];  const float* cb0 = (const float*)d_in[5];
  const float*     dw0 = (const float*)d_in[6];  const float* db0 = (const float*)d_in[7];
  const float*     cw1 = (const float*)d_in[8];  const float* cb1 = (const float*)d_in[9];
  const float*     dw1 = (const float*)d_in[10]; const float* db1 = (const float*)d_in[11];
  const float*     cw2 = (const float*)d_in[12]; const float* cb2 = (const float*)d_in[13];
  const float*     dw2 = (const float*)d_in[14]; const float* db2 = (const float*)d_in[15];
  const float*     fcw = (const float*)d_in[16]; const float* fcb = (const float*)d_in[17];
  float* out = (float*)d_out;

  float* ws = (float*)d_ws;
  size_t o = 0;
  float* deg  = ws + o; o += MTOT;
  float* h    = ws + o; o += (size_t)MTOT * 12;
  float* X0   = ws + o; o += (size_t)MTOT * 12;
  float* wt0c = ws + o; o += (size_t)3 * 12 * 128;
  float* wt0d = ws + o; o += (size_t)12 * 128;
  float* wt1c = ws + o; o += (size_t)3 * 128 * 512;
  float* wt1d = ws + o; o += (size_t)128 * 512;
  float* wt2c = ws + o; o += (size_t)3 * 512 * 128;
  float* wt2d = ws + o; o += (size_t)512 * 128;
  float* fcwT = ws + o; o += (size_t)3840 * 80;
  float* X1   = ws + o; o += (size_t)MTOT * 128;   // block0 out; reused as X3
  float* X2   = ws + o; o += (size_t)MTOT * 512;   // block1 out
  float* X3   = X1;

  const int TPB = 256;
  // --- GCN
  gcn_deg_init <<<(MTOT + TPB - 1) / TPB, TPB, 0, stream>>>(deg);
  gcn_deg_count<<<(ECNT + TPB - 1) / TPB, TPB, 0, stream>>>(ei, deg);
  gcn_dinv     <<<(MTOT + TPB - 1) / TPB, TPB, 0, stream>>>(deg);
  gcn_h_kernel <<<(MTOT + TPB - 1) / TPB, TPB, 0, stream>>>(x, gcn_w, h);
  gcn_self     <<<(MTOT * 12 + TPB - 1) / TPB, TPB, 0, stream>>>(h, deg, gcn_b, X0);
  {
    long long tot = (long long)ECNT * 12;
    gcn_scatter<<<(unsigned)((tot + TPB - 1) / TPB), TPB, 0, stream>>>(ei, h, deg, X0);
  }
  // --- weight layout prep
  transpose_conv_w<<<(128 * 12 * 3  + TPB - 1) / TPB, TPB, 0, stream>>>(cw0, wt0c, 128, 12);
  transpose_down_w<<<(128 * 12      + TPB - 1) / TPB, TPB, 0, stream>>>(dw0, wt0d, 128, 12);
  transpose_conv_w<<<(512 * 128 * 3 + TPB - 1) / TPB, TPB, 0, stream>>>(cw1, wt1c, 512, 128);
  transpose_down_w<<<(512 * 128     + TPB - 1) / TPB, TPB, 0, stream>>>(dw1, wt1d, 512, 128);
  transpose_conv_w<<<(128 * 512 * 3 + TPB - 1) / TPB, TPB, 0, stream>>>(cw2, wt2c, 128, 512);
  transpose_down_w<<<(128 * 512     + TPB - 1) / TPB, TPB, 0, stream>>>(dw2, wt2d, 128, 512);
  transpose_fc_w  <<<(3840 * 80     + TPB - 1) / TPB, TPB, 0, stream>>>(fcw, fcwT);

  // --- fused TCN blocks (WMMA f32 + async-LDS staged panels)
  tcn_block<12,  128, 1><<<dim3(960, 2), 128, 0, stream>>>(X0, wt0c, cb0, wt0d, db0, X1);
  tcn_block<128, 512, 3><<<dim3(960, 8), 128, 0, stream>>>(X1, wt1c, cb1, wt1d, db1, X2);
  tcn_block<512, 128, 9><<<dim3(960, 2), 128, 0, stream>>>(X2, wt2c, cb2, wt2d, db2, X3);

  // --- FC
  fc_kernel<5><<<dim3(128, 1), 160, 0, stream>>>(X3, fcwT, fcb, out);
}